// SpatialGraphAttention_44152263803157
// MI455X (gfx1250) — compile-verified
//
#include <hip/hip_runtime.h>
#include <hip/hip_bf16.h>
#include <math.h>

typedef float v2f __attribute__((ext_vector_type(2)));
typedef float v8f __attribute__((ext_vector_type(8)));

#define IN_CH 256
#define OUT_CH 64
#define HEADS 4
#define NEG_SLOPE 0.2f

// ---- monotonic float<->uint mapping for atomicMax on signed floats ----
__device__ __forceinline__ unsigned f2ord(float f) {
    unsigned u = __float_as_uint(f);
    return (u & 0x80000000u) ? ~u : (u | 0x80000000u);
}
__device__ __forceinline__ float ord2f(unsigned t) {
    return (t & 0x80000000u) ? __uint_as_float(t & 0x7fffffffu)
                             : __uint_as_float(~t);
}

// ---------------- generic fill (init accumulators / -inf sentinels) -------
__global__ void fill_u32(unsigned* __restrict__ p, unsigned v, size_t n) {
    size_t i = (size_t)blockIdx.x * blockDim.x + threadIdx.x;
    size_t stride = (size_t)gridDim.x * blockDim.x;
    for (; i < n; i += stride) p[i] = v;
}

// ---------------- f32 WMMA GEMM: C[M,Ncol] = A[M,K] @ B[K,Ncol] -----------
// One wave32 computes one 16x16 tile with V_WMMA_F32_16X16X4_F32.
// A (16x4 f32): lane = (K>=2)*16 + M, VGPR v holds K = 2*(lane/16)+v.
// B (4x16 f32): lane = (K>=2)*16 + N, VGPR v holds K = 2*(lane/16)+v.
// C/D (16x16 f32): VGPR v: lanes 0-15 -> M=v, lanes 16-31 -> M=8+v; N=lane&15.
__global__ void gemm_wmma_f32(const float* __restrict__ A,
                              const float* __restrict__ B,
                              float* __restrict__ C,
                              int M, int K, int Ncol) {
    int wave = (int)(((size_t)blockIdx.x * blockDim.x + threadIdx.x) >> 5);
    int lane = threadIdx.x & 31;
    int tilesN = Ncol >> 4;
    int tilesM = M >> 4;
    if (wave >= tilesM * tilesN) return;
    int tm = wave / tilesN;
    int tn = wave - tm * tilesN;

    int mrow = (tm << 4) + (lane & 15);
    int ncol = (tn << 4) + (lane & 15);
    int kgrp = (lane >> 4) << 1;          // 0 or 2

    const float* Arow = A + (size_t)mrow * K;
    v8f acc = {0.f, 0.f, 0.f, 0.f, 0.f, 0.f, 0.f, 0.f};

    for (int k0 = 0; k0 < K; k0 += 4) {
        v2f a, b;
        a.x = Arow[k0 + kgrp];
        a.y = Arow[k0 + kgrp + 1];
        b.x = B[(size_t)(k0 + kgrp) * Ncol + ncol];
        b.y = B[(size_t)(k0 + kgrp + 1) * Ncol + ncol];
        acc = __builtin_amdgcn_wmma_f32_16x16x4_f32(
            false, a, false, b, (short)0, acc, false, false);
    }

    int rbase = (tm << 4) + ((lane >> 4) << 3);
    float* Cp = C + (size_t)rbase * Ncol + (tn << 4) + (lane & 15);
#pragma unroll
    for (int v = 0; v < 8; ++v)
        Cp[(size_t)v * Ncol] = acc[v];
}

// ------------- attention scores: a_s/a_d[n,h] = <h[n,h,:], att[h,:]> ------
__global__ void att_scores(const float* __restrict__ h,
                           const float* __restrict__ att_s,
                           const float* __restrict__ att_d,
                           float* __restrict__ a_s, float* __restrict__ a_d,
                           int N, int H, int C) {
    int idx = blockIdx.x * blockDim.x + threadIdx.x;
    if (idx >= N * H) return;
    int hd = idx % H;
    const float* hp = h + (size_t)idx * C;   // [N,H,C] row-major; idx = n*H+hd
    const float* as = att_s + hd * C;
    const float* ad = att_d + hd * C;
    float s = 0.f, d = 0.f;
    for (int c = 0; c < C; ++c) {
        float v = hp[c];
        s = fmaf(v, as[c], s);
        d = fmaf(v, ad[c], d);
    }
    a_s[idx] = s;
    a_d[idx] = d;
}

// ------------- edge helpers ----------------------------------------------
__device__ __forceinline__ void edge_sd(const long long* ei, int E, int e,
                                        int& s, int& d) {
    if (e < E) { s = (int)ei[e]; d = (int)ei[E + e]; }
    else       { s = e - E; d = s; }               // appended self loops
}

template <int H>
__global__ void edge_alpha_max(const long long* __restrict__ ei, int E, int N,
                               const float* __restrict__ a_s,
                               const float* __restrict__ a_d,
                               unsigned* __restrict__ mord) {
    int e = blockIdx.x * blockDim.x + threadIdx.x;
    if (e >= E + N) return;
    int s, d;
    edge_sd(ei, E, e, s, d);
#pragma unroll
    for (int h = 0; h < H; ++h) {
        float al = a_s[s * H + h] + a_d[d * H + h];
        al = (al >= 0.f) ? al : al * NEG_SLOPE;    // leaky_relu
        atomicMax(&mord[d * H + h], f2ord(al));
    }
}

template <int H>
__global__ void edge_exp_sum(const long long* __restrict__ ei, int E, int N,
                             const float* __restrict__ a_s,
                             const float* __restrict__ a_d,
                             const unsigned* __restrict__ mord,
                             float* __restrict__ denom,
                             float* __restrict__ ebuf) {
    int e = blockIdx.x * blockDim.x + threadIdx.x;
    if (e >= E + N) return;
    int s, d;
    edge_sd(ei, E, e, s, d);
#pragma unroll
    for (int h = 0; h < H; ++h) {
        float al = a_s[s * H + h] + a_d[d * H + h];
        al = (al >= 0.f) ? al : al * NEG_SLOPE;
        float m = ord2f(mord[d * H + h]);
        float ex = expf(al - m);
        ebuf[(size_t)e * H + h] = ex;
        atomicAdd(&denom[d * H + h], ex);
    }
}

// one wave32 per edge; coalesced 32-lane strips; f32 atomic scatter-add
template <int H, int C>
__global__ void edge_accum(const long long* __restrict__ ei, int E, int N,
                           const float* __restrict__ hfeat,
                           const float* __restrict__ denom,
                           const float* __restrict__ ebuf,
                           float* __restrict__ out) {
    size_t gtid = (size_t)blockIdx.x * blockDim.x + threadIdx.x;
    int e = (int)(gtid >> 5);
    int lane = threadIdx.x & 31;
    if (e >= E + N) return;
    int s, d;
    edge_sd(ei, E, e, s, d);
    float coef[H];
#pragma unroll
    for (int h = 0; h < H; ++h)
        coef[h] = ebuf[(size_t)e * H + h] / denom[d * H + h];
    const float* hp = hfeat + (size_t)s * (H * C);
    float* op = out + (size_t)d * (H * C);
#pragma unroll
    for (int i = 0; i < (H * C) / 32; ++i) {
        int idx = lane + (i << 5);
        atomicAdd(&op[idx], hp[idx] * coef[idx / C]);
    }
}

// ------------- epilogues --------------------------------------------------
__global__ void bias_elu_inplace(float* __restrict__ p,
                                 const float* __restrict__ bias,
                                 int colmask, size_t n) {
    size_t i = (size_t)blockIdx.x * blockDim.x + threadIdx.x;
    size_t stride = (size_t)gridDim.x * blockDim.x;
    for (; i < n; i += stride) {
        float v = p[i] + bias[i & colmask];
        p[i] = (v > 0.f) ? v : expm1f(v);
    }
}

__global__ void bias_add(const float* __restrict__ in,
                         const float* __restrict__ bias,
                         float* __restrict__ out, int colmask, size_t n) {
    size_t i = (size_t)blockIdx.x * blockDim.x + threadIdx.x;
    size_t stride = (size_t)gridDim.x * blockDim.x;
    for (; i < n; i += stride) out[i] = in[i] + bias[i & colmask];
}

// ------------- host orchestration ----------------------------------------
static inline int cdiv_i(long long a, long long b) { return (int)((a + b - 1) / b); }

extern "C" void kernel_launch(void* const* d_in, const int* in_sizes, int n_in,
                              void* d_out, int out_size, void* d_ws, size_t ws_size,
                              hipStream_t stream) {
    (void)n_in; (void)out_size; (void)ws_size;
    const float*     x   = (const float*)d_in[0];
    const long long* ei  = (const long long*)d_in[1];   // int64 edge_index [2,E]
    const float*     W1  = (const float*)d_in[2];
    const float*     as1 = (const float*)d_in[3];
    const float*     ad1 = (const float*)d_in[4];
    const float*     b1  = (const float*)d_in[5];
    const float*     W2  = (const float*)d_in[6];
    const float*     as2 = (const float*)d_in[7];
    const float*     ad2 = (const float*)d_in[8];
    const float*     b2  = (const float*)d_in[9];
    float* out = (float*)d_out;

    const int N  = in_sizes[0] / IN_CH;   // 30000
    const int E  = in_sizes[1] / 2;       // 480000
    const int ET = E + N;                 // with self loops
    const int F1 = HEADS * OUT_CH;        // 256

    // ---- workspace carving (floats); layer2 aliases dead layer1 space ----
    float* ws = (float*)d_ws;
    size_t NF1 = (size_t)N * F1;
    size_t NH  = (size_t)N * HEADS;
    size_t NO  = (size_t)N * OUT_CH;
    float*    h1    = ws;                        // [N,256]
    float*    acc1  = ws + NF1;                  // [N,256] accum -> h2 input
    float*    small = ws + 2 * NF1;
    float*    asrc1 = small;                     // N*H
    float*    adst1 = small + NH;                // N*H
    unsigned* mord1 = (unsigned*)(small + 2 * NH);
    float*    den1  = small + 3 * NH;
    float*    ebuf1 = small + 4 * NH;            // ET*H
    float*    h2f   = h1;                        // [N,64]
    float*    acc2  = h1 + NO;                   // [N,64]
    float*    asrc2 = h1 + 2 * NO;               // N
    float*    adst2 = asrc2 + N;                 // N
    unsigned* mord2 = (unsigned*)(adst2 + N);    // N
    float*    den2  = adst2 + 2 * N;             // N
    float*    ebuf2 = adst2 + 3 * N;             // ET

    const int B = 256;

    fill_u32<<<cdiv_i(NF1, B), B, 0, stream>>>((unsigned*)acc1, 0u, NF1);
    fill_u32<<<cdiv_i(2 * NH, B), B, 0, stream>>>(mord1, 0u, 2 * NH); // mord1+den1

    // ---- layer 1 ----
    {
        long long waves = (long long)(N >> 4) * (F1 >> 4);
        gemm_wmma_f32<<<cdiv_i(waves * 32, B), B, 0, stream>>>(x, W1, h1, N, IN_CH, F1);
    }
    att_scores<<<cdiv_i((long long)N * HEADS, B), B, 0, stream>>>(
        h1, as1, ad1, asrc1, adst1, N, HEADS, OUT_CH);
    edge_alpha_max<HEADS><<<cdiv_i(ET, B), B, 0, stream>>>(ei, E, N, asrc1, adst1, mord1);
    edge_exp_sum<HEADS><<<cdiv_i(ET, B), B, 0, stream>>>(ei, E, N, asrc1, adst1,
                                                         mord1, den1, ebuf1);
    edge_accum<HEADS, OUT_CH><<<cdiv_i((long long)ET * 32, B), B, 0, stream>>>(
        ei, E, N, h1, den1, ebuf1, acc1);
    bias_elu_inplace<<<cdiv_i(NF1, B), B, 0, stream>>>(acc1, b1, F1 - 1, NF1);

    // ---- layer 2 ----
    fill_u32<<<cdiv_i(NO, B), B, 0, stream>>>((unsigned*)acc2, 0u, NO);
    fill_u32<<<cdiv_i(2 * (long long)N, B), B, 0, stream>>>(mord2, 0u, 2 * (size_t)N);
    {
        long long waves = (long long)(N >> 4) * (OUT_CH >> 4);
        gemm_wmma_f32<<<cdiv_i(waves * 32, B), B, 0, stream>>>(acc1, W2, h2f, N, F1, OUT_CH);
    }
    att_scores<<<cdiv_i(N, B), B, 0, stream>>>(h2f, as2, ad2, asrc2, adst2, N, 1, OUT_CH);
    edge_alpha_max<1><<<cdiv_i(ET, B), B, 0, stream>>>(ei, E, N, asrc2, adst2, mord2);
    edge_exp_sum<1><<<cdiv_i(ET, B), B, 0, stream>>>(ei, E, N, asrc2, adst2,
                                                     mord2, den2, ebuf2);
    edge_accum<1, OUT_CH><<<cdiv_i((long long)ET * 32, B), B, 0, stream>>>(
        ei, E, N, h2f, den2, ebuf2, acc2);
    bias_add<<<cdiv_i(NO, B), B, 0, stream>>>(acc2, b2, out, OUT_CH - 1, NO);
}